// GalacticGNN_5299989643769
// MI455X (gfx1250) — compile-verified
//
#include <hip/hip_runtime.h>
#include <hip/hip_bf16.h>

// ---------------------------------------------------------------------------
// GCN (3x GCNConv + GraphNorm + ReLU) + FC head for MI455X (gfx1250, wave32)
// N=100000 nodes, E=1600000 edges, IN=HID=64, OUT=32.
// Edge scatter/gather dominates (~3.3GB traffic, L2-resident at 192MB);
// dense GEMMs run on the fp32 matrix pipe (v_wmma_f32_16x16x4_f32).
// ---------------------------------------------------------------------------

#define GN_N        100000
#define GN_E        1600000
#define GN_HID      64
#define GN_OUT      32
#define GN_EPS      1e-5f

typedef __attribute__((ext_vector_type(2)))  float v2f;
typedef __attribute__((ext_vector_type(8)))  float v8f;

// ---------------------------------------------------------------------------
// Degree: deg[dst] += 1 for each edge (deg buffer pre-zeroed)
// ---------------------------------------------------------------------------
__global__ void gnn_deg_scatter(const long long* __restrict__ ei, float* __restrict__ deg, int E) {
    int e = blockIdx.x * blockDim.x + threadIdx.x;
    if (e >= E) return;
    int dst = (int)ei[(long long)E + e];
    atomicAdd(&deg[dst], 1.0f);
}

// dinv = rsqrt(deg + 1) in place
__global__ void gnn_deg_finalize(float* __restrict__ d, int n) {
    int i = blockIdx.x * blockDim.x + threadIdx.x;
    if (i >= n) return;
    d[i] = __frsqrt_rn(d[i] + 1.0f);
}

// ---------------------------------------------------------------------------
// WMMA GEMM: out[nrows x NCOLS] = in[nrows x 64] @ W[64 x NCOLS] (+ bias)
// 256 threads = 8 wave32's; each wave owns one 16x16 tile.
// Compile-time NCOLS -> constant LDS strides -> ds_load with imm offsets.
// All 16 A-frags + 16 B-frags preloaded to VGPRs, then 16 back-to-back
// v_wmma_f32_16x16x4_f32 on two independent accumulator chains.
// ---------------------------------------------------------------------------
template <int NCOLS>
__global__ void gnn_wmma_gemm(const float* __restrict__ in, const float* __restrict__ W,
                              const float* __restrict__ bias, float* __restrict__ out,
                              int nrows) {
    constexpr int COLT   = NCOLS / 16;   // 4 (hid) or 2 (fc)
    constexpr int ROWT   = 8 / COLT;     // 2 or 4
    constexpr int ROWSPB = ROWT * 16;    // 32 or 64

    __shared__ float sW[64 * NCOLS];
    __shared__ float sX[ROWSPB * 64];

    const int tid     = threadIdx.x;
    const int baseRow = blockIdx.x * ROWSPB;

    // Stage W (64 x NCOLS) into LDS
    #pragma unroll
    for (int i = tid * 4; i < 64 * NCOLS; i += 256 * 4)
        *(float4*)&sW[i] = *(const float4*)&W[i];

    // Stage X tile (ROWSPB x 64) into LDS, clamp out-of-range rows
    #pragma unroll
    for (int i = tid * 4; i < ROWSPB * 64; i += 256 * 4) {
        int r  = baseRow + (i >> 6);
        int c  = i & 63;
        int rr = r < nrows ? r : (nrows - 1);
        *(float4*)&sX[i] = *(const float4*)&in[(long long)rr * 64 + c];
    }
    __syncthreads();

    const int wave = tid >> 5;
    const int lane = tid & 31;
    const int ct   = wave % COLT;
    const int rt   = wave / COLT;
    const int half = lane >> 4;          // selects K pair {0,1} vs {2,3} per step
    const int l15  = lane & 15;

    const int arow = (rt * 16 + l15) * 64;     // A: row base in sX
    const int bcol = ct * 16 + l15;            // B: column in sW

    // Preload all fragments (A: 16x64 -> 32 VGPRs; B: 64x16 -> 32 VGPRs)
    v2f afr[16], bfr[16];
    #pragma unroll
    for (int k4 = 0; k4 < 16; ++k4) {
        const int kb = k4 * 4 + half * 2;
        afr[k4][0] = sX[arow + kb];
        afr[k4][1] = sX[arow + kb + 1];
        bfr[k4][0] = sW[kb * NCOLS + bcol];
        bfr[k4][1] = sW[(kb + 1) * NCOLS + bcol];
    }

    // Two independent accumulator chains to shorten the D->C dependency
    v8f acc0 = {}, acc1 = {};
    #pragma unroll
    for (int k4 = 0; k4 < 16; k4 += 2) {
        acc0 = __builtin_amdgcn_wmma_f32_16x16x4_f32(false, afr[k4], false, bfr[k4],
                                                     (short)0, acc0, false, false);
        acc1 = __builtin_amdgcn_wmma_f32_16x16x4_f32(false, afr[k4 + 1], false, bfr[k4 + 1],
                                                     (short)0, acc1, false, false);
    }

    // C/D layout: VGPR i -> lanes 0-15: M=i, lanes 16-31: M=i+8; N = lane&15
    const float bv = bias ? bias[bcol] : 0.0f;
    #pragma unroll
    for (int i = 0; i < 8; ++i) {
        int grow = baseRow + rt * 16 + half * 8 + i;
        if (grow < nrows)
            out[(long long)grow * NCOLS + bcol] = acc0[i] + acc1[i] + bv;
    }
}

// ---------------------------------------------------------------------------
// out = h * dinv[row]^2 + b[f]   (self-loop term + bias, init for scatter)
// ---------------------------------------------------------------------------
__global__ void gnn_selfloop_init(const float* __restrict__ h, const float* __restrict__ dinv,
                                  const float* __restrict__ b, float* __restrict__ out, int n) {
    int i = blockIdx.x * blockDim.x + threadIdx.x;
    if (i >= n) return;
    int row = i >> 6, f = i & 63;
    float d = dinv[row];
    out[i] = h[i] * d * d + b[f];
}

// ---------------------------------------------------------------------------
// Edge aggregation: thread = (edge, 4-feature chunk); 16 lanes cover one
// 256B feature row -> coalesced gather, contiguous atomics on dst row.
// ---------------------------------------------------------------------------
__global__ void gnn_edge_scatter(const float* __restrict__ h, const long long* __restrict__ ei,
                                 const float* __restrict__ dinv, float* __restrict__ out, int E) {
    long long tid = (long long)blockIdx.x * blockDim.x + threadIdx.x;
    int e = (int)(tid >> 4);
    int c = (int)(tid & 15);
    if (e >= E) return;
    __builtin_prefetch(&ei[e + 2048], 0, 0);
    int src = (int)ei[e];
    int dst = (int)ei[(long long)E + e];
    float nrm = dinv[src] * dinv[dst];
    float4 v = *(const float4*)&h[(long long)src * 64 + c * 4];
    float* o = out + (long long)dst * 64 + c * 4;
    atomicAdd(o + 0, v.x * nrm);
    atomicAdd(o + 1, v.y * nrm);
    atomicAdd(o + 2, v.z * nrm);
    atomicAdd(o + 3, v.w * nrm);
}

// ---------------------------------------------------------------------------
// GraphNorm pass 1: per-feature column sums (block: 128 rows, LDS reduce,
// one atomicAdd per feature per block)
// ---------------------------------------------------------------------------
__global__ void gnn_colsum(const float* __restrict__ h, float* __restrict__ csum, int nrows) {
    __shared__ float red[256];
    int f = threadIdx.x & 63, rg = threadIdx.x >> 6;
    int base = blockIdx.x * 128;
    float acc = 0.0f;
    for (int k = rg; k < 128; k += 4) {
        int r = base + k;
        if (r < nrows) acc += h[(long long)r * 64 + f];
    }
    red[threadIdx.x] = acc;
    __syncthreads();
    if (threadIdx.x < 64)
        atomicAdd(&csum[f], red[f] + red[64 + f] + red[128 + f] + red[192 + f]);
}

// pass 2: t = h - mean*gm (in place), accumulate per-feature sum(t^2)
__global__ void gnn_center_sumsq(float* __restrict__ h, const float* __restrict__ csum,
                                 float* __restrict__ csq, const float* __restrict__ gm,
                                 int nrows, float invN) {
    __shared__ float red[256];
    int f = threadIdx.x & 63, rg = threadIdx.x >> 6;
    int base = blockIdx.x * 128;
    float mshift = csum[f] * invN * gm[f];
    float acc = 0.0f;
    for (int k = rg; k < 128; k += 4) {
        int r = base + k;
        if (r < nrows) {
            long long idx = (long long)r * 64 + f;
            float t = h[idx] - mshift;
            h[idx] = t;
            acc += t * t;
        }
    }
    red[threadIdx.x] = acc;
    __syncthreads();
    if (threadIdx.x < 64)
        atomicAdd(&csq[f], red[f] + red[64 + f] + red[128 + f] + red[192 + f]);
}

// pass 3: h = relu(gw * t * rsqrt(var + eps) + gb)
__global__ void gnn_norm_relu(float* __restrict__ h, const float* __restrict__ csq,
                              const float* __restrict__ gw, const float* __restrict__ gb,
                              int n, float invN) {
    int i = blockIdx.x * blockDim.x + threadIdx.x;
    if (i >= n) return;
    int f = i & 63;
    float var = csq[f] * invN;
    float v = gw[f] * h[i] * __frsqrt_rn(var + GN_EPS) + gb[f];
    h[i] = v > 0.0f ? v : 0.0f;
}

// ---------------------------------------------------------------------------
// Host-side launch
// ---------------------------------------------------------------------------
extern "C" void kernel_launch(void* const* d_in, const int* in_sizes, int n_in,
                              void* d_out, int out_size, void* d_ws, size_t ws_size,
                              hipStream_t stream) {
    (void)in_sizes; (void)n_in; (void)out_size; (void)ws_size;

    const float*     x  = (const float*)d_in[0];
    const long long* ei = (const long long*)d_in[1];
    // per layer: W, b, gw, gb, gm at indices 2+5*i .. 6+5*i
    const float* fcW = (const float*)d_in[17];
    const float* fcb = (const float*)d_in[18];
    float* out = (float*)d_out;

    // Workspace layout (floats): pong[N*64] | ping[N*64] | dinv[N] | csum[64] | csq[64]
    float* pong = (float*)d_ws;
    float* ping = pong + (size_t)GN_N * GN_HID;
    float* dinv = ping + (size_t)GN_N * GN_HID;
    float* csum = dinv + GN_N;
    float* csq  = csum + 64;

    const float invN = 1.0f / (float)GN_N;

    // Degree -> deg_inv_sqrt
    hipMemsetAsync(dinv, 0, (size_t)GN_N * sizeof(float), stream);
    gnn_deg_scatter<<<(GN_E + 255) / 256, 256, 0, stream>>>(ei, dinv, GN_E);
    gnn_deg_finalize<<<(GN_N + 255) / 256, 256, 0, stream>>>(dinv, GN_N);

    const float* cur = x;  // layer input
    for (int L = 0; L < 3; ++L) {
        const float* W  = (const float*)d_in[2 + 5 * L];
        const float* b  = (const float*)d_in[3 + 5 * L];
        const float* gw = (const float*)d_in[4 + 5 * L];
        const float* gb = (const float*)d_in[5 + 5 * L];
        const float* gm = (const float*)d_in[6 + 5 * L];

        // h = cur @ W   (pong)
        gnn_wmma_gemm<GN_HID><<<(GN_N + 31) / 32, 256, 0, stream>>>(cur, W, nullptr, pong, GN_N);
        // ping = h * dii + b, then scatter-add messages
        gnn_selfloop_init<<<(GN_N * GN_HID + 255) / 256, 256, 0, stream>>>(pong, dinv, b, ping,
                                                                           GN_N * GN_HID);
        gnn_edge_scatter<<<(unsigned)(((long long)GN_E * 16 + 255) / 256), 256, 0, stream>>>(
            pong, ei, dinv, ping, GN_E);
        // GraphNorm + ReLU (in place on ping)
        hipMemsetAsync(csum, 0, 128 * sizeof(float), stream);
        gnn_colsum<<<(GN_N + 127) / 128, 256, 0, stream>>>(ping, csum, GN_N);
        gnn_center_sumsq<<<(GN_N + 127) / 128, 256, 0, stream>>>(ping, csum, csq, gm, GN_N, invN);
        gnn_norm_relu<<<(GN_N * GN_HID + 255) / 256, 256, 0, stream>>>(ping, csq, gw, gb,
                                                                       GN_N * GN_HID, invN);
        cur = ping;
    }

    // Final FC: out = ping @ fcW + fcb
    gnn_wmma_gemm<GN_OUT><<<(GN_N + 63) / 64, 256, 0, stream>>>(cur, fcW, fcb, out, GN_N);
}